// QuantumPoolingLayer_4312147165377
// MI455X (gfx1250) — compile-verified
//
#include <hip/hip_runtime.h>
#include <hip/hip_bf16.h>

// ---------------------------------------------------------------------------
// QuantumPoolingLayer, MI455X (gfx1250, wave32)
//
// Stage A (dominant): stream 537 MB of inputs once, quadrant-sum reduction.
//   HBM roofline: 537MB / 23.3 TB/s ~= 23 us. Implemented as a fused kernel:
//   grid = 64 batches x 32 m-tiles (2048 blocks, 256 thr = 8 waves).
//   Each wave owns (quadrant, row-half) of one (b, 16-wide m-tile) and
//   reduces 512 rows x 16 cols with V_WMMA_F32_16X16X4_F32 using an
//   all-ones A matrix: D[i][j] = sum_k B[k][j] + C[i][j]  (exact f32 adds,
//   independent of the precise A/B VGPR row layout). 128 WMMAs / wave,
//   256 dword loads with compile-time immediate offsets for deep MLP.
// Stage B (negligible): 16 threads/block run tanh + the 4-qubit circuit
//   (16 complex amps in registers, statically indexed -> no scratch).
// ---------------------------------------------------------------------------

typedef __attribute__((ext_vector_type(2))) float v2f;
typedef __attribute__((ext_vector_type(8))) float v8f;

#define N_BATCH 64
#define N_M     512

// Apply 2x2 complex gate [[a,b],[c,d]] on the qubit selected by MASK (bit of
// the 4-bit basis index; wire w has MASK = 8 >> w, wire 0 = MSB).
template<int MASK>
__device__ __forceinline__ void apply_gate(float re[16], float im[16],
    float ar, float ai, float br, float bi,
    float cr, float ci, float dr, float di) {
#pragma unroll
  for (int i0 = 0; i0 < 16; ++i0) {
    if (i0 & MASK) continue;
    const int i1 = i0 | MASK;
    const float xr = re[i0], xi = im[i0];
    const float yr = re[i1], yi = im[i1];
    re[i0] = ar * xr - ai * xi + br * yr - bi * yi;
    im[i0] = ar * xi + ai * xr + br * yi + bi * yr;
    re[i1] = cr * xr - ci * xi + dr * yr - di * yi;
    im[i1] = cr * xi + ci * xr + dr * yi + di * yr;
  }
}

template<int CMASK, int TMASK>
__device__ __forceinline__ void cnot_swap(float re[16], float im[16]) {
#pragma unroll
  for (int k = 0; k < 16; ++k) {
    if ((k & CMASK) && !(k & TMASK)) {
      const int k2 = k | TMASK;
      float t;
      t = re[k]; re[k] = re[k2]; re[k2] = t;
      t = im[k]; im[k] = im[k2]; im[k2] = t;
    }
  }
}

// PennyLane Rot(phi,theta,omega) = RZ(om) RY(th) RZ(phi)
__device__ __forceinline__ void rot_params(float phi, float th, float om,
    float& ar, float& ai, float& br, float& bi,
    float& cr, float& ci, float& dr, float& di) {
  float sp, cp, sq, cq, st, ct;
  sincosf(0.5f * (phi + om), &sp, &cp);
  sincosf(0.5f * (phi - om), &sq, &cq);
  sincosf(0.5f * th, &st, &ct);
  ar =  ct * cp;  ai = -ct * sp;   // exp(-i(phi+om)/2) * cos(th/2)
  br = -st * cq;  bi = -st * sq;   // -exp(+i(phi-om)/2) * sin(th/2)
  cr =  st * cq;  ci = -st * sq;   //  exp(-i(phi-om)/2) * sin(th/2)
  dr =  ct * cp;  di =  ct * sp;   //  exp(+i(phi+om)/2) * cos(th/2)
}

#define RY_APPLY(MASK, xq) do {                                   \
    float s_, c_;                                                 \
    sincosf((xq) * 1.5707963267948966f, &s_, &c_);                \
    apply_gate<MASK>(re, im, c_, 0.f, -s_, 0.f, s_, 0.f, c_, 0.f);\
  } while (0)

#define ROT_APPLY(MASK, base) do {                                \
    float ar_, ai_, br_, bi_, cr_, ci_, dr_, di_;                 \
    rot_params(wp[(base) + 0], wp[(base) + 1], wp[(base) + 2],    \
               ar_, ai_, br_, bi_, cr_, ci_, dr_, di_);           \
    apply_gate<MASK>(re, im, ar_, ai_, br_, bi_, cr_, ci_, dr_, di_); \
  } while (0)

__global__ __launch_bounds__(256)
void qpool_fused_kernel(const float* __restrict__ in,
                        const float* __restrict__ qw,
                        float* __restrict__ out) {
  __shared__ float colsum[8][16];

  const int tid  = threadIdx.x;
  const int w    = tid >> 5;          // wave id 0..7
  const int lane = tid & 31;
  const int col  = lane & 15;         // WMMA column = m within tile
  const int hsel = lane >> 4;         // lane half -> which 2 of 4 B-rows

  const int b  = blockIdx.x >> 5;          // batch 0..63
  const int m0 = (blockIdx.x & 31) << 4;   // m-tile origin (16-wide)

  const int quad  = w >> 1;           // 0..3 : (p,q) quadrant, row-major
  const int rhalf = w & 1;            // which 512 of the 1024 quadrant rows
  const int r0 = (quad >> 1) * 32;
  const int c0 = (quad & 1) * 32;
  // element offset of (b, r, c, m) = ((b*64 + r)*64 + c)*512 + m
  const size_t rowbase = (size_t)b * 4096 + (size_t)r0 * 64 + (size_t)c0;

#if __has_builtin(__builtin_amdgcn_wmma_f32_16x16x4_f32)
  v8f acc = {0.f, 0.f, 0.f, 0.f, 0.f, 0.f, 0.f, 0.f};
  const v2f aone = {1.0f, 1.0f};      // all-ones A => D[:,j] = sum_k B[k][j]
  for (int o = 0; o < 16; ++o) {      // 16 image rows per wave (of 32)
    const int rline = rhalf * 16 + o; // row within quadrant / 32-col line
    const float* p = in + ((rowbase + (size_t)rline * 64 + 2 * hsel) * 512
                           + (size_t)m0 + (size_t)col);
#pragma unroll
    for (int i = 0; i < 8; ++i) {     // 8 x (4 rows x 16 cols) per line
      v2f bv;
      bv[0] = p[i * 2048];            // imm offsets: i*8192B (+2048B) < 2^23
      bv[1] = p[i * 2048 + 512];
      acc = __builtin_amdgcn_wmma_f32_16x16x4_f32(
          false, aone, false, bv, (short)0, acc, false, false);
    }
  }
  if (lane < 16) colsum[w][col] = acc[0];   // every D row == column sums
#else
  float a0 = 0.f, a1 = 0.f;
  for (int o = 0; o < 16; ++o) {
    const int rline = rhalf * 16 + o;
    const float* p = in + ((rowbase + (size_t)rline * 64 + 2 * hsel) * 512
                           + (size_t)m0 + (size_t)col);
#pragma unroll
    for (int i = 0; i < 8; ++i) {
      a0 += p[i * 2048];
      a1 += p[i * 2048 + 512];
    }
  }
  float accs = a0 + a1;
  accs += __shfl_xor(accs, 16, 32);   // combine the two lane-halves
  if (lane < 16) colsum[w][col] = accs;
#endif

  __syncthreads();

  // ---- Stage B: tanh + 4-qubit circuit, 16 samples per block -------------
  if (tid < 16) {
    const int m = m0 + tid;
    float x[4];
#pragma unroll
    for (int q = 0; q < 4; ++q) {
      const float s = colsum[2 * q][tid] + colsum[2 * q + 1][tid];
      x[q] = tanhf(s * (1.0f / 1024.0f));   // mean over 32x32 quadrant
    }

    float re[16], im[16];
#pragma unroll
    for (int k = 0; k < 16; ++k) { re[k] = 0.f; im[k] = 0.f; }
    re[0] = 1.f;

    // Initial RY(x[q]*pi) embedding, wire q (MSB-first bit masks 8,4,2,1)
    RY_APPLY(8, x[0]);
    RY_APPLY(4, x[1]);
    RY_APPLY(2, x[2]);
    RY_APPLY(1, x[3]);

    // weights index: flat sample idx = b*512 + m ; widx = idx // 64
    const int widx = b * 8 + (m >> 6);
    const float* wp = qw + (size_t)widx * 24;   // [2][4][3]

#pragma unroll
    for (int layer = 0; layer < 2; ++layer) {
      const int lb = layer * 12;
      ROT_APPLY(8, lb + 0);
      ROT_APPLY(4, lb + 3);
      ROT_APPLY(2, lb + 6);
      ROT_APPLY(1, lb + 9);
      // CNOT(i,j) for all i<j, in order
      cnot_swap<8, 4>(re, im);
      cnot_swap<8, 2>(re, im);
      cnot_swap<8, 1>(re, im);
      cnot_swap<4, 2>(re, im);
      cnot_swap<4, 1>(re, im);
      cnot_swap<2, 1>(re, im);
    }

    float p0 = 0.f, p1 = 0.f;
#pragma unroll
    for (int k = 0; k < 8; ++k)  p0 += re[k] * re[k] + im[k] * im[k];
#pragma unroll
    for (int k = 8; k < 16; ++k) p1 += re[k] * re[k] + im[k] * im[k];

    out[(size_t)b * N_M + m] = p0 - p1;   // <Z0>
  }
}

extern "C" void kernel_launch(void* const* d_in, const int* in_sizes, int n_in,
                              void* d_out, int out_size, void* d_ws, size_t ws_size,
                              hipStream_t stream) {
  (void)in_sizes; (void)n_in; (void)out_size; (void)d_ws; (void)ws_size;
  const float* inputs = (const float*)d_in[0];   // [64,64,64,512] f32
  const float* qw     = (const float*)d_in[1];   // [512,2,4,3]   f32
  float* out          = (float*)d_out;           // [64,512]      f32

  // 64 batches x 32 m-tiles = 2048 blocks, 256 threads (8 waves) each.
  qpool_fused_kernel<<<dim3(N_BATCH * 32), dim3(256), 0, stream>>>(inputs, qw, out);
}